// GaussianScatterAndAvg3D_59055800320120
// MI455X (gfx1250) — compile-verified
//
#include <hip/hip_runtime.h>
#include <hip/hip_bf16.h>

// Problem constants (from reference)
#define Bn 4
#define Sn 2048
#define Rn 64
#define Cn 16
#define Pn 5
#define P3n 125
#define NPOINTS (Bn * Sn)                 // 8192
#define NVOX (Bn * Rn * Rn * Rn)          // 1,048,576 voxel-rows
#define GRID_CH 17                        // 16 feature channels + 1 count channel
#define GWS_FLOATS ((size_t)NVOX * GRID_CH)

typedef float v2f __attribute__((ext_vector_type(2)));
typedef float v8f __attribute__((ext_vector_type(8)));

// ---------------------------------------------------------------------------
// Kernel 0: zero the grid workspace (grid-stride)
// ---------------------------------------------------------------------------
__global__ void zero_ws_kernel(float* __restrict__ gws, size_t n) {
    size_t i = (size_t)blockIdx.x * blockDim.x + threadIdx.x;
    size_t stride = (size_t)gridDim.x * blockDim.x;
    for (; i < n; i += stride) gws[i] = 0.0f;
}

// ---------------------------------------------------------------------------
// Kernel 1: per-(point, offset) scatter.
// One block per point (128 threads, 125 active = one offset each).
// Replicates the reference's crossed feature/grid channel mapping:
//   for flat I in [0,2000): off = I/16, feat_ch = I%16, grid_ch = I/125
// counts only tracked for grid_ch==0 (I<125 -> offsets 0..7, weights 16..16,13)
// ---------------------------------------------------------------------------
__global__ void scatter_kernel(const float* __restrict__ positions,
                               const float* __restrict__ x,
                               const float* __restrict__ w_poca,
                               const float* __restrict__ b_poca,
                               float* __restrict__ gws) {
    const int p = blockIdx.x;          // point index 0..B*S-1
    const int j = threadIdx.x;         // offset index 0..124
    if (j >= P3n) return;

    const float* posp = positions + (size_t)p * 3;
    const float* xp   = x + (size_t)p * Cn;

    // load x into registers (reused 16x below)
    float xv[Cn];
#pragma unroll
    for (int c = 0; c < Cn; ++c) xv[c] = xp[c];

    // corrections = 0.1 * sigmoid(x @ w_poca + b_poca); pos = (positions+corr)*R
    float pos[3];
#pragma unroll
    for (int d = 0; d < 3; ++d) {
        float t = b_poca[d];
#pragma unroll
        for (int c = 0; c < Cn; ++c) t += xv[c] * w_poca[c * 3 + d];
        float s = 1.0f / (1.0f + expf(-t));
        pos[d] = (posp[d] + 0.1f * s) * (float)Rn;
    }

    const int ox = j / 25 - 2;
    const int oy = (j / 5) % 5 - 2;
    const int oz = j % 5 - 2;

    // frac = pos - rint(pos)  (round half to even, matches jnp.rint)
    const float dx = (pos[0] - rintf(pos[0])) + (float)ox;
    const float dy = (pos[1] - rintf(pos[1])) + (float)oy;
    const float dz = (pos[2] - rintf(pos[2])) + (float)oz;

    const float STD = 0.6f / 64.0f;
    const float inv = 1.0f / (STD * STD);
    const float g = expf(-(dx * dx + dy * dy + dz * dz) * inv);

    // voxel = clip(int(pos) + off, 0, R-1)  (truncation; pos >= 0 here)
    int vx = (int)pos[0] + ox; vx = vx < 0 ? 0 : (vx > Rn - 1 ? Rn - 1 : vx);
    int vy = (int)pos[1] + oy; vy = vy < 0 ? 0 : (vy > Rn - 1 ? Rn - 1 : vy);
    int vz = (int)pos[2] + oz; vz = vz < 0 ? 0 : (vz > Rn - 1 ? Rn - 1 : vz);

    const int b = p >> 11;  // S = 2048
    const size_t row = (size_t)b * (Rn * Rn * Rn) + ((size_t)vx << 12) + ((size_t)vy << 6) + (size_t)vz;
    float* cell = gws + row * GRID_CH;

    // feature scatter: value = x[I%16] * g into grid channel I/125, I = 16*j + t.
    // g==+0.0 for the vast majority of offsets (exp underflow); adding (+/-)0.0
    // to a (+0-initialized, monotone-accumulated) cell is a value-identity, skip.
    if (g != 0.0f) {
#pragma unroll
        for (int t = 0; t < Cn; ++t) {
            const int gc = (Cn * j + t) / P3n;   // grid channel (crossed mapping)
            atomicAdd(cell + gc, xv[t] * g);
        }
    }

    // counts for grid channel 0 live at channel index 16
    const int cnt = (j < 7) ? 16 : ((j == 7) ? 13 : 0);
    if (cnt) atomicAdd(cell + 16, (float)cnt);
}

// ---------------------------------------------------------------------------
// Kernel 2: out = grid17 @ w_out + b_out via V_WMMA_F32_16X16X4_F32.
// One wave32 per 16-row x 16-col tile; K=17 padded to 20 -> 5 WMMA issues.
// f32 16x4 A layout: lanes 0-15 hold (M=lane, K=k0+{0,1}),
//                    lanes 16-31 hold (M=lane-16, K=k0+{2,3}); B mirrors on K.
// C/D layout: VGPR j -> row (half*8 + j), col = lane%16.
// ---------------------------------------------------------------------------
__global__ void out_gemm_kernel(const float* __restrict__ gws,
                                const float* __restrict__ w_out,
                                const float* __restrict__ b_out,
                                float* __restrict__ out) {
    const int wave = blockIdx.x * (blockDim.x >> 5) + (threadIdx.x >> 5);
    const int lane = threadIdx.x & 31;
    const int row0 = wave * 16;          // wave-uniform; NVOX divisible by 16

    const int half = lane >> 4;          // 0 or 1
    const int l16  = lane & 15;
    const int arow = row0 + l16;

    v8f acc = {};
#pragma unroll
    for (int k0 = 0; k0 < 20; k0 += 4) {
        const int ka = k0 + half * 2;
        v2f a, bm;
        a.x  = (ka + 0 < GRID_CH) ? gws[(size_t)arow * GRID_CH + ka + 0] : 0.0f;
        a.y  = (ka + 1 < GRID_CH) ? gws[(size_t)arow * GRID_CH + ka + 1] : 0.0f;
        bm.x = (ka + 0 < GRID_CH) ? w_out[(ka + 0) * Cn + l16] : 0.0f;
        bm.y = (ka + 1 < GRID_CH) ? w_out[(ka + 1) * Cn + l16] : 0.0f;
        // 8 args: (neg_a, A, neg_b, B, c_mod, C, reuse_a, reuse_b)
        acc = __builtin_amdgcn_wmma_f32_16x16x4_f32(
            false, a, false, bm, (short)0, acc, false, false);
    }

    const float bias = b_out[l16];
#pragma unroll
    for (int jv = 0; jv < 8; ++jv) {
        const size_t orow = (size_t)(row0 + half * 8 + jv);
        out[orow * Cn + l16] = acc[jv] + bias;
    }
}

// ---------------------------------------------------------------------------
// Host-side launcher
// ---------------------------------------------------------------------------
extern "C" void kernel_launch(void* const* d_in, const int* in_sizes, int n_in,
                              void* d_out, int out_size, void* d_ws, size_t ws_size,
                              hipStream_t stream) {
    const float* positions = (const float*)d_in[0];  // (B,S,3)
    const float* x         = (const float*)d_in[1];  // (B,S,C)
    const float* w_poca    = (const float*)d_in[2];  // (C,3)
    const float* b_poca    = (const float*)d_in[3];  // (3,)
    const float* w_out     = (const float*)d_in[4];  // (C+1,C)
    const float* b_out     = (const float*)d_in[5];  // (C,)
    float* out = (float*)d_out;                      // (B,R,R,R,C)
    float* gws = (float*)d_ws;                       // (NVOX, 17) scratch

    // 1) zero grid workspace
    zero_ws_kernel<<<4096, 256, 0, stream>>>(gws, GWS_FLOATS);

    // 2) scatter: one block per point, one thread per offset
    scatter_kernel<<<NPOINTS, 128, 0, stream>>>(positions, x, w_poca, b_poca, gws);

    // 3) output GEMM: NVOX/16 = 65536 tiles, 8 waves (256 thr) per block
    const int tiles = NVOX / 16;                     // 65536
    const int waves_per_block = 256 / 32;            // 8
    out_gemm_kernel<<<tiles / waves_per_block, 256, 0, stream>>>(gws, w_out, b_out, out);
}